// PerfectFeatureModel_77618648973643
// MI455X (gfx1250) — compile-verified
//
#include <hip/hip_runtime.h>
#include <cmath>

// ---------------------------------------------------------------------------
// Problem constants (from the reference)
// ---------------------------------------------------------------------------
#define LRN     1024    // low-res nodes
#define HRN     2048    // high-res nodes
#define F1      4096    // conv1 out features (8 heads x 512)
#define F2      2048    // conv2 out features (4 heads x 512)
#define D_HEAD  512
#define EPSGN   1e-5f

typedef float v2f __attribute__((ext_vector_type(2)));
typedef float v8f __attribute__((ext_vector_type(8)));

// V_WMMA_F32_16X16X4_F32 : D(16x16 f32) = A(16x4 f32) * B(4x16 f32) + C
// 8-arg form: (neg_a, A, neg_b, B, c_mod, C, reuse_a, reuse_b)
__device__ __forceinline__ v8f wmma4(v2f a, v2f b, v8f c) {
  return __builtin_amdgcn_wmma_f32_16x16x4_f32(false, a, false, b, (short)0, c,
                                               false, false);
}

// ---------------------------------------------------------------------------
// WMMA GEMM (NN): C[M,N] = A[M,K] @ B[K,N] (+bias) (+=C if accumulate)
// grid = (N/128, M/128), block = 128 threads (4 waves, 2x2 of 64x64 wave
// tiles; each wave holds a 4x4 grid of 16x16 f32 accumulators = 128 VGPRs).
// ---------------------------------------------------------------------------
__global__ __launch_bounds__(128)
void wmma_gemm_nn(const float* __restrict__ A, int lda,
                  const float* __restrict__ B, int ldb,
                  const float* __restrict__ bias,
                  float* __restrict__ C, int ldc,
                  int K, int accumulate) {
  const int lane = threadIdx.x & 31;
  const int wave = threadIdx.x >> 5;
  const int half = lane >> 4;         // selects K-pair (ISA A-layout)
  const int r    = lane & 15;
  const int m0   = blockIdx.y * 128 + (wave >> 1) * 64;
  const int n0   = blockIdx.x * 128 + (wave & 1) * 64;

  const v8f vzero = {0.f,0.f,0.f,0.f,0.f,0.f,0.f,0.f};
  v8f acc[4][4];
#pragma unroll
  for (int mi = 0; mi < 4; ++mi)
#pragma unroll
    for (int ni = 0; ni < 4; ++ni) acc[mi][ni] = vzero;

  const float* Ar[4];
#pragma unroll
  for (int mi = 0; mi < 4; ++mi)
    Ar[mi] = A + (size_t)(m0 + 16 * mi + r) * lda;

#pragma unroll 2
  for (int k0 = 0; k0 < K; k0 += 4) {
    const int k = k0 + 2 * half;
    v2f a[4], b[4];
#pragma unroll
    for (int mi = 0; mi < 4; ++mi) a[mi] = *(const v2f*)(Ar[mi] + k);
    const float* Bk0 = B + (size_t)k * ldb + n0 + r;
    const float* Bk1 = Bk0 + ldb;
#pragma unroll
    for (int ni = 0; ni < 4; ++ni) {
      b[ni].x = Bk0[ni * 16];
      b[ni].y = Bk1[ni * 16];
    }
#pragma unroll
    for (int mi = 0; mi < 4; ++mi)
#pragma unroll
      for (int ni = 0; ni < 4; ++ni)
        acc[mi][ni] = wmma4(a[mi], b[ni], acc[mi][ni]);
  }

#pragma unroll
  for (int mi = 0; mi < 4; ++mi)
#pragma unroll
    for (int ni = 0; ni < 4; ++ni) {
      const int colg = n0 + ni * 16 + r;
      const float bv = bias ? bias[colg] : 0.f;
#pragma unroll
      for (int v = 0; v < 8; ++v) {
        const int rowg = m0 + mi * 16 + v + 8 * half;   // ISA C-layout
        const size_t idx = (size_t)rowg * ldc + colg;
        float val = acc[mi][ni][v] + bv;
        if (accumulate) val += C[idx];
        C[idx] = val;
      }
    }
}

// ---------------------------------------------------------------------------
// WMMA GEMM (NT): C[M,N] = f(scale * (A[M,K] @ B[N,K]^T + qe[i]*e[i,j]))
// e[i,j] = eX[j*eLD + i]   (edge features = x^T in layer 1)
// f = ReLU if do_relu.  grid = (N/128, M/128), block = 128.
// ---------------------------------------------------------------------------
__global__ __launch_bounds__(128)
void wmma_gemm_nt(const float* __restrict__ A, int lda,
                  const float* __restrict__ B, int ldb,
                  float* __restrict__ C, int ldc,
                  int K, float scale, int do_relu,
                  const float* __restrict__ qe,
                  const float* __restrict__ eX, int eLD) {
  const int lane = threadIdx.x & 31;
  const int wave = threadIdx.x >> 5;
  const int half = lane >> 4;
  const int r    = lane & 15;
  const int m0   = blockIdx.y * 128 + (wave >> 1) * 64;
  const int n0   = blockIdx.x * 128 + (wave & 1) * 64;

  const v8f vzero = {0.f,0.f,0.f,0.f,0.f,0.f,0.f,0.f};
  v8f acc[4][4];
#pragma unroll
  for (int mi = 0; mi < 4; ++mi)
#pragma unroll
    for (int ni = 0; ni < 4; ++ni) acc[mi][ni] = vzero;

  const float* Ar[4];
  const float* Br[4];
#pragma unroll
  for (int t = 0; t < 4; ++t) {
    Ar[t] = A + (size_t)(m0 + 16 * t + r) * lda;
    Br[t] = B + (size_t)(n0 + 16 * t + r) * ldb;
  }

#pragma unroll 2
  for (int k0 = 0; k0 < K; k0 += 4) {
    const int k = k0 + 2 * half;
    v2f a[4], b[4];
#pragma unroll
    for (int t = 0; t < 4; ++t) {
      a[t] = *(const v2f*)(Ar[t] + k);
      b[t] = *(const v2f*)(Br[t] + k);
    }
#pragma unroll
    for (int mi = 0; mi < 4; ++mi)
#pragma unroll
      for (int ni = 0; ni < 4; ++ni)
        acc[mi][ni] = wmma4(a[mi], b[ni], acc[mi][ni]);
  }

#pragma unroll
  for (int mi = 0; mi < 4; ++mi)
#pragma unroll
    for (int ni = 0; ni < 4; ++ni) {
      const int colg = n0 + ni * 16 + r;
#pragma unroll
      for (int v = 0; v < 8; ++v) {
        const int rowg = m0 + mi * 16 + v + 8 * half;
        float val = acc[mi][ni][v];
        if (qe) val += qe[rowg] * eX[(size_t)colg * eLD + rowg];
        val *= scale;
        if (do_relu) val = fmaxf(val, 0.f);
        C[(size_t)rowg * ldc + colg] = val;
      }
    }
}

// ---------------------------------------------------------------------------
// Row softmax, in place.  grid = nrows, block = 256.
// ---------------------------------------------------------------------------
__global__ __launch_bounds__(256)
void softmax_rows(float* __restrict__ S, int ncols) {
  __shared__ float red[256];
  const int tid = threadIdx.x;
  float* Sr = S + (size_t)blockIdx.x * ncols;

  float m = -3.0e38f;
  for (int j = tid; j < ncols; j += 256) m = fmaxf(m, Sr[j]);
  red[tid] = m; __syncthreads();
  for (int s = 128; s > 0; s >>= 1) {
    if (tid < s) red[tid] = fmaxf(red[tid], red[tid + s]);
    __syncthreads();
  }
  m = red[0]; __syncthreads();

  float sum = 0.f;
  for (int j = tid; j < ncols; j += 256) {
    float e = expf(Sr[j] - m);
    Sr[j] = e;
    sum += e;
  }
  red[tid] = sum; __syncthreads();
  for (int s = 128; s > 0; s >>= 1) {
    if (tid < s) red[tid] += red[tid + s];
    __syncthreads();
  }
  const float inv = 1.f / red[0];
  for (int j = tid; j < ncols; j += 256) Sr[j] *= inv;
}

// qe[h*LRN + i] = dot(Q[i, h*512 : h*512+512], We[h*512 : ...])
__global__ void qe_kernel(const float* __restrict__ Q,
                          const float* __restrict__ We,
                          float* __restrict__ qe) {
  const int t = blockIdx.x * 256 + threadIdx.x;   // 8*LRN threads
  const int h = t >> 10, i = t & (LRN - 1);
  const float* qr = Q + (size_t)i * F1 + h * D_HEAD;
  const float* w  = We + h * D_HEAD;
  float s = 0.f;
  for (int d = 0; d < D_HEAD; ++d) s += qr[d] * w[d];
  qe[t] = s;
}

// ae[i] = sum_j alpha[i,j] * x[j,i]     (layer-1 edge aggregation, per head)
__global__ __launch_bounds__(256)
void ae_kernel(const float* __restrict__ S, const float* __restrict__ x,
               float* __restrict__ ae) {
  __shared__ float red[256];
  const int i = blockIdx.x, tid = threadIdx.x;
  float s = 0.f;
  for (int j = tid; j < LRN; j += 256)
    s += S[(size_t)i * LRN + j] * x[(size_t)j * LRN + i];
  red[tid] = s; __syncthreads();
  for (int k = 128; k > 0; k >>= 1) {
    if (tid < k) red[tid] += red[tid + k];
    __syncthreads();
  }
  if (tid == 0) ae[i] = red[0];
}

// H[i, off+d] += ae[i] * We[off+d]
__global__ void edge_out_kernel(float* __restrict__ H,
                                const float* __restrict__ ae,
                                const float* __restrict__ We, int off) {
  const int t = blockIdx.x * 256 + threadIdx.x;   // LRN * D_HEAD threads
  const int i = t >> 9, d = t & (D_HEAD - 1);
  H[(size_t)i * F1 + off + d] += ae[i] * We[off + d];
}

// GraphNorm stats over node dim (rows)
__global__ void col_mean_kernel(const float* __restrict__ H,
                                float* __restrict__ mean, int M, int F) {
  const int f = blockIdx.x * 256 + threadIdx.x;
  if (f >= F) return;
  float s = 0.f;
  for (int i = 0; i < M; ++i) s += H[(size_t)i * F + f];
  mean[f] = s / (float)M;
}

__global__ void col_var_kernel(const float* __restrict__ H,
                               const float* __restrict__ mean,
                               const float* __restrict__ ms,
                               float* __restrict__ var, int M, int F) {
  const int f = blockIdx.x * 256 + threadIdx.x;
  if (f >= F) return;
  const float mu = ms[f] * mean[f];
  float s = 0.f;
  for (int i = 0; i < M; ++i) {
    const float t = H[(size_t)i * F + f] - mu;
    s += t * t;
  }
  var[f] = s / (float)M;
}

__global__ void gn_apply_kernel(float* __restrict__ H,
                                const float* __restrict__ mean,
                                const float* __restrict__ var,
                                const float* __restrict__ ms,
                                const float* __restrict__ g,
                                const float* __restrict__ b, int Fmask) {
  const size_t t = (size_t)blockIdx.x * 256 + threadIdx.x;
  const int f = (int)(t & (size_t)Fmask);
  const float mu = ms[f] * mean[f];
  H[t] = g[f] * (H[t] - mu) * rsqrtf(var[f] + EPSGN) + b[f];
}

// Row L2 normalize.  grid = nrows, block = 256.
__global__ __launch_bounds__(256)
void l2norm_rows_kernel(float* __restrict__ H, int F) {
  __shared__ float red[256];
  const int tid = threadIdx.x;
  float* Hr = H + (size_t)blockIdx.x * F;
  float s = 0.f;
  for (int j = tid; j < F; j += 256) { const float v = Hr[j]; s += v * v; }
  red[tid] = s; __syncthreads();
  for (int k = 128; k > 0; k >>= 1) {
    if (tid < k) red[tid] += red[tid + k];
    __syncthreads();
  }
  const float inv = 1.f / sqrtf(red[0]);
  for (int j = tid; j < F; j += 256) Hr[j] *= inv;
}

// out[c, r] = in[r, c].  grid = (C/32, R/32), block = 256 (32x8).
__global__ __launch_bounds__(256)
void transpose_kernel(const float* __restrict__ in, float* __restrict__ out,
                      int R, int Ccols) {
  __shared__ float tile[32][33];
  const int bx = blockIdx.x * 32, by = blockIdx.y * 32;
  const int tx = threadIdx.x & 31, ty = threadIdx.x >> 5;
  for (int yy = ty; yy < 32; yy += 8)
    tile[yy][tx] = in[(size_t)(by + yy) * Ccols + (bx + tx)];
  __syncthreads();
  for (int yy = ty; yy < 32; yy += 8)
    out[(size_t)(bx + yy) * R + (by + tx)] = tile[tx][yy];
}

// ---------------------------------------------------------------------------
// Host orchestration
// ---------------------------------------------------------------------------
extern "C" void kernel_launch(void* const* d_in, const int* in_sizes, int n_in,
                              void* d_out, int out_size, void* d_ws,
                              size_t ws_size, hipStream_t stream) {
  const float* x      = (const float*)d_in[0];
  const float* Wq1    = (const float*)d_in[1];
  const float* bq1    = (const float*)d_in[2];
  const float* Wk1    = (const float*)d_in[3];
  const float* bk1    = (const float*)d_in[4];
  const float* Wv1    = (const float*)d_in[5];
  const float* bv1    = (const float*)d_in[6];
  const float* We1    = (const float*)d_in[7];
  const float* Wskip1 = (const float*)d_in[8];
  const float* bskip1 = (const float*)d_in[9];
  const float* g1     = (const float*)d_in[10];
  const float* bb1    = (const float*)d_in[11];
  const float* ms1    = (const float*)d_in[12];
  const float* Wq2    = (const float*)d_in[13];
  const float* bq2    = (const float*)d_in[14];
  const float* Wk2    = (const float*)d_in[15];
  const float* bk2    = (const float*)d_in[16];
  const float* Wv2    = (const float*)d_in[17];
  const float* bv2    = (const float*)d_in[18];
  const float* Wskip2 = (const float*)d_in[19];
  const float* bskip2 = (const float*)d_in[20];
  const float* g2     = (const float*)d_in[21];
  const float* bb2    = (const float*)d_in[22];
  const float* ms2    = (const float*)d_in[23];

  float* out_hr = (float*)d_out;                      // [2048, 2048]
  float* out_lr = out_hr + (size_t)HRN * HRN;         // [1024, 1024]

  // Workspace layout (units of 1M floats).  Peak ~60M floats = 240 MB.
  float* ws = (float*)d_ws;
  const size_t MF = 1ull << 20;
  float* Q1  = ws;              // 4M   [1024,4096]
  float* K1b = ws + 4  * MF;    // 4M
  float* V1  = ws + 8  * MF;    // 4M
  float* H1  = ws + 12 * MF;    // 4M   skip -> +attn -> GN -> lr_x
  float* S1  = ws + 16 * MF;    // 1M   per-head scores [1024,1024]
  float* XT  = ws + 24 * MF;    // 4M   lr_x^T  [4096,1024]
  float* V2  = ws + 28 * MF;    // 8M
  float* H2  = ws + 36 * MF;    // 8M   skip2 -> +attn -> GN
  float* Q2  = ws;              // 8M   (aliases Q1/K1b, dead by then)
  float* K2  = ws + 16 * MF;    // 8M   (aliases S1 reserve region)
  float* S2  = ws + 44 * MF;    // 16M  per-head scores [4096,4096]
  float* HRX = ws;              // 8M   [2048,4096] (aliases Q2, dead by then)
  float* mean = ws + 60 * MF;   // 4096
  float* var  = mean + 4096;    // 4096
  float* qe   = var + 4096;     // 8*1024
  float* ae   = qe + 8 * 1024;  // 1024
  (void)in_sizes; (void)n_in; (void)out_size; (void)ws_size;

  const dim3 blk(128);
  const float scale = 1.0f / sqrtf((float)D_HEAD);

  // ---- conv1 projections:  [1024,1024] @ [1024,4096] ----
  wmma_gemm_nn<<<dim3(F1/128, LRN/128), blk, 0, stream>>>(x, LRN, Wq1, F1, bq1,    Q1,  F1, LRN, 0);
  wmma_gemm_nn<<<dim3(F1/128, LRN/128), blk, 0, stream>>>(x, LRN, Wk1, F1, bk1,    K1b, F1, LRN, 0);
  wmma_gemm_nn<<<dim3(F1/128, LRN/128), blk, 0, stream>>>(x, LRN, Wv1, F1, bv1,    V1,  F1, LRN, 0);
  wmma_gemm_nn<<<dim3(F1/128, LRN/128), blk, 0, stream>>>(x, LRN, Wskip1, F1, bskip1, H1, F1, LRN, 0);
  qe_kernel<<<(8 * LRN) / 256, 256, 0, stream>>>(Q1, We1, qe);

  // ---- conv1 attention, per head (8 heads, D=512) ----
  for (int h = 0; h < 8; ++h) {
    wmma_gemm_nt<<<dim3(LRN/128, LRN/128), blk, 0, stream>>>(
        Q1 + h * D_HEAD, F1, K1b + h * D_HEAD, F1, S1, LRN, D_HEAD,
        scale, 0, qe + h * LRN, x, LRN);
    softmax_rows<<<LRN, 256, 0, stream>>>(S1, LRN);
    ae_kernel<<<LRN, 256, 0, stream>>>(S1, x, ae);
    wmma_gemm_nn<<<dim3(D_HEAD/128, LRN/128), blk, 0, stream>>>(
        S1, LRN, V1 + h * D_HEAD, F1, nullptr, H1 + h * D_HEAD, F1, LRN, 1);
    edge_out_kernel<<<(LRN * D_HEAD) / 256, 256, 0, stream>>>(H1, ae, We1, h * D_HEAD);
  }

  // ---- GraphNorm1 + row L2 norm -> lr_x (in H1) ----
  col_mean_kernel<<<F1 / 256, 256, 0, stream>>>(H1, mean, LRN, F1);
  col_var_kernel <<<F1 / 256, 256, 0, stream>>>(H1, mean, ms1, var, LRN, F1);
  gn_apply_kernel<<<(LRN * (size_t)F1) / 256, 256, 0, stream>>>(H1, mean, var, ms1, g1, bb1, F1 - 1);
  l2norm_rows_kernel<<<LRN, 256, 0, stream>>>(H1, F1);

  // ---- lr output: relu(lr_x @ lr_x^T) ----
  wmma_gemm_nt<<<dim3(LRN/128, LRN/128), blk, 0, stream>>>(
      H1, F1, H1, F1, out_lr, LRN, F1, 1.0f, 1, nullptr, nullptr, 0);

  // ---- conv2: nodes are columns of lr_x ----
  transpose_kernel<<<dim3(F1/32, LRN/32), 256, 0, stream>>>(H1, XT, LRN, F1);
  wmma_gemm_nn<<<dim3(F2/128, F1/128), blk, 0, stream>>>(XT, LRN, Wq2, F2, bq2,    Q2, F2, LRN, 0);
  wmma_gemm_nn<<<dim3(F2/128, F1/128), blk, 0, stream>>>(XT, LRN, Wk2, F2, bk2,    K2, F2, LRN, 0);
  wmma_gemm_nn<<<dim3(F2/128, F1/128), blk, 0, stream>>>(XT, LRN, Wv2, F2, bv2,    V2, F2, LRN, 0);
  wmma_gemm_nn<<<dim3(F2/128, F1/128), blk, 0, stream>>>(XT, LRN, Wskip2, F2, bskip2, H2, F2, LRN, 0);

  // ---- conv2 attention, per head (4 heads, N=4096, D=512) ----
  for (int h = 0; h < 4; ++h) {
    wmma_gemm_nt<<<dim3(F1/128, F1/128), blk, 0, stream>>>(
        Q2 + h * D_HEAD, F2, K2 + h * D_HEAD, F2, S2, F1, D_HEAD,
        scale, 0, nullptr, nullptr, 0);
    softmax_rows<<<F1, 256, 0, stream>>>(S2, F1);
    wmma_gemm_nn<<<dim3(D_HEAD/128, F1/128), blk, 0, stream>>>(
        S2, F1, V2 + h * D_HEAD, F2, nullptr, H2 + h * D_HEAD, F2, F1, 1);
  }

  // ---- GraphNorm2, transpose -> hr_x [2048,4096], L2 norm ----
  col_mean_kernel<<<F2 / 256, 256, 0, stream>>>(H2, mean, F1, F2);
  col_var_kernel <<<F2 / 256, 256, 0, stream>>>(H2, mean, ms2, var, F1, F2);
  gn_apply_kernel<<<((size_t)F1 * F2) / 256, 256, 0, stream>>>(H2, mean, var, ms2, g2, bb2, F2 - 1);
  transpose_kernel<<<dim3(F2/32, F1/32), 256, 0, stream>>>(H2, HRX, F1, F2);
  l2norm_rows_kernel<<<HRN, 256, 0, stream>>>(HRX, F1);

  // ---- hr output: relu(hr_x @ hr_x^T) ----
  wmma_gemm_nt<<<dim3(HRN/128, HRN/128), blk, 0, stream>>>(
      HRX, F1, HRX, F1, out_hr, HRN, F1, 1.0f, 1, nullptr, nullptr, 0);
}